// ContinuallyLearningPrototypes_63084479644188
// MI455X (gfx1250) — compile-verified
//
#include <hip/hip_runtime.h>
#include <hip/hip_bf16.h>

typedef __attribute__((ext_vector_type(2))) float v2f;
typedef __attribute__((ext_vector_type(8))) float v8f;

#define N_SAMPLES 32768
#define DIM       64
#define NPROTO    4096
#define NCLASS    100

// One workgroup = 256 threads = 8 waves; each wave owns 16 samples (columns).
// The workgroup streams all 4096 prototypes through a 32-row LDS tile and each
// wave accumulates two independent 16x16 fp32 WMMA chains over K=64.
__global__ __launch_bounds__(256)
void clp_fused_kernel(const float* __restrict__ X,
                      const float* __restrict__ protos,
                      const float* __restrict__ simth,
                      const int*   __restrict__ labels,
                      float* __restrict__ out)
{
    // 68-float row stride => A-fragment ds_load_b64 across the wave hits all
    // 64 LDS banks exactly once (bank stride 4 per M row).
    __shared__ float        protoTile[32][68];          // 8704 B
    __shared__ unsigned int scores[8][16][NCLASS];      // 51200 B (per-wave regions)

    const int tid  = threadIdx.x;
    const int w    = tid >> 5;       // wave in workgroup
    const int lane = tid & 31;
    const int col  = lane & 15;      // N index within 16x16 tile
    const int hi   = lane >> 4;      // half-wave select

    // init per-wave class-max tables to 0 (== 0.0f, matches reference's zeros)
    for (int i = tid; i < 8 * 16 * NCLASS; i += 256)
        ((unsigned int*)scores)[i] = 0u;

    const int n0   = blockIdx.x * 128 + w * 16;  // first sample of this wave
    const int nrow = n0 + col;

    // Load B fragments (Xn^T tiles) for the whole K=64 and compute 1/||X[n]||.
    // Lane L holds B[k][col] = X[nrow][k] for k = 4*ks + 2*hi + {0,1}.
    v2f bfrag[16];
    float ss = 0.f;
    #pragma unroll
    for (int ks = 0; ks < 16; ++ks) {
        const int k = 4 * ks + 2 * hi;
        v2f b = *(const v2f*)&X[(size_t)nrow * DIM + k];
        bfrag[ks] = b;
        ss += b.x * b.x + b.y * b.y;
    }
    // lanes L and L^16 hold complementary halves of the same row
    ss += __shfl_xor(ss, 16, 32);
    float inv = rsqrtf(ss);
    inv = inv * (1.5f - 0.5f * ss * inv * inv);   // one Newton step -> fp32 accurate

    __syncthreads();   // scores init + before first tile overwrite

    for (int m0 = 0; m0 < NPROTO; m0 += 32) {
        // cooperative coalesced load: 32 prototype rows (8 KB) into LDS
        #pragma unroll
        for (int i = 0; i < 2; ++i) {
            const int idx = tid + i * 256;       // 0..511 float4 slots
            const int row = idx >> 4;            // 0..31
            const int k4  = (idx & 15) << 2;     // 0,4,...,60
            const float4 pv = *(const float4*)&protos[(size_t)(m0 + row) * DIM + k4];
            *(float4*)&protoTile[row][k4] = pv;
        }
        __syncthreads();

        // Two independent accumulator chains (rows m0..m0+15 and m0+16..m0+31)
        v8f acc0 = {};
        v8f acc1 = {};
        #pragma unroll
        for (int ks = 0; ks < 16; ++ks) {
            const int bk = 4 * ks + 2 * hi;
            v2f a0 = *(const v2f*)&protoTile[col][bk];
            v2f a1 = *(const v2f*)&protoTile[col + 16][bk];
            acc0 = __builtin_amdgcn_wmma_f32_16x16x4_f32(false, a0, false, bfrag[ks],
                                                         (short)0, acc0, false, false);
            acc1 = __builtin_amdgcn_wmma_f32_16x16x4_f32(false, a1, false, bfrag[ks],
                                                         (short)0, acc1, false, false);
        }

        // Threshold + per-class running max. D VGPR v at lane L is
        // sims[m0 + v + 8*hi][n0 + col] (chain 0) / +16 (chain 1).
        // Kept values are >= sim_th (0.45) > 0, so uint atomicMax == float max.
        #pragma unroll
        for (int v = 0; v < 8; ++v) {
            const int mA = m0 + v + 8 * hi;
            const float s0 = acc0[v] * inv;
            if (s0 >= simth[mA]) {
                const int lbl = labels[mA];
                if ((unsigned)lbl < (unsigned)NCLASS)
                    atomicMax(&scores[w][col][lbl], __float_as_uint(s0));
            }
            const int mB = mA + 16;
            const float s1 = acc1[v] * inv;
            if (s1 >= simth[mB]) {
                const int lbl = labels[mB];
                if ((unsigned)lbl < (unsigned)NCLASS)
                    atomicMax(&scores[w][col][lbl], __float_as_uint(s1));
            }
        }
        __syncthreads();   // tile reads done before next overwrite
    }

    // Each wave writes its 16 samples x 100 classes (row-major N x C output).
    for (int s = 0; s < 16; ++s) {
        const unsigned int* src = scores[w][s];
        float* dst = &out[(size_t)(n0 + s) * NCLASS];
        for (int c = lane; c < NCLASS; c += 32)
            dst[c] = __uint_as_float(src[c]);
    }
}

extern "C" void kernel_launch(void* const* d_in, const int* in_sizes, int n_in,
                              void* d_out, int out_size, void* d_ws, size_t ws_size,
                              hipStream_t stream) {
    (void)in_sizes; (void)n_in; (void)out_size; (void)d_ws; (void)ws_size;
    const float* X      = (const float*)d_in[0];
    const float* protos = (const float*)d_in[1];
    const float* simth  = (const float*)d_in[2];
    const int*   labels = (const int*)d_in[3];
    float*       out    = (float*)d_out;

    dim3 grid(N_SAMPLES / 128);   // 256 workgroups
    dim3 block(256);              // 8 waves
    hipLaunchKernelGGL(clp_fused_kernel, grid, block, 0, stream,
                       X, protos, simth, labels, out);
}